// LINK_4501125726311
// MI455X (gfx1250) — compile-verified
//
#include <hip/hip_runtime.h>
#include <hip/hip_bf16.h>

// LINK forward: out[i,:] = b[:] + sum over edges (i->j) of W[:, j]
// N = 100000 nodes, OUT = 64 channels, E = 3.2M edges.
//
//   K1: transpose W [64,N] -> WT [N,64] in d_ws (L2-resident, 25.6 MB)
//   K2: init out[i,:] = b[:] using v_wmma_f32_16x16x4_f32 (exact f32 math),
//       full 16-row tiles only -> guard-free clause stores; scalar tail kernel
//       for n%16 rows (not launched when n%16==0).
//   K3: per-edge wave32 scatter: coalesced 256B read of WT[col],
//       64x global_atomic_add_f32 into out[row]; global_prefetch_b8 ahead;
//       NT loads on the once-streamed edge index array.

typedef __attribute__((ext_vector_type(2))) float v2f;
typedef __attribute__((ext_vector_type(8))) float v8f;

#define OUTC 64

// ---------------------------------------------------------------------------
// K1: W[o, j] (row-major [OUT, N]) -> WT[j, o] ([N, OUT]) via LDS 64x64 tiles
// ---------------------------------------------------------------------------
__global__ void __launch_bounds__(256)
link_transpose_w(const float* __restrict__ W, float* __restrict__ WT, int n) {
    __shared__ float tile[64][65];              // +1 pad: conflict-free both ways
    const int j0 = blockIdx.x * 64;
    const int tx = threadIdx.x & 63;
    const int ty = threadIdx.x >> 6;            // 0..3

#pragma unroll
    for (int rr = 0; rr < 16; ++rr) {
        const int o = ty + rr * 4;              // row of W (channel), 0..63
        const int j = j0 + tx;
        tile[o][tx] = (j < n) ? W[(size_t)o * (size_t)n + (size_t)j] : 0.0f;
    }
    __syncthreads();
#pragma unroll
    for (int rr = 0; rr < 16; ++rr) {
        const int jj = ty + rr * 4;             // node index within tile
        const int j  = j0 + jj;
        if (j < n) WT[(size_t)j * OUTC + tx] = tile[tx][jj];
    }
}

// ---------------------------------------------------------------------------
// K2: out[i, c] = b[c] broadcast via WMMA f32 16x16x4 over FULL 16-row tiles.
//     A = 0.25 everywhere, B column n = b[n] (all k) =>
//     D[m,n] = 4 * 0.25 * b[n] = b[n]  (f32 FMA chain, <=1 ulp).
//     One wave covers 16 rows x 64 cols with 4 WMMAs; unconditional
//     constant-offset stores (callers guarantee ntiles*16 <= nrows).
// ---------------------------------------------------------------------------
__global__ void __launch_bounds__(256)
link_init_bias_wmma(const float* __restrict__ b, float* __restrict__ out, int ntiles) {
    const int wave = (int)((blockIdx.x * blockDim.x + threadIdx.x) >> 5);
    const int lane = (int)(threadIdx.x & 31);
    if (wave >= ntiles) return;                 // wave-uniform: EXEC all-1 inside

    const int nsub  = lane & 15;                // column within 16-wide tile
    const int rbase = wave * 16 + ((lane >> 4) << 3); // lanes 16..31: M = v + 8

    v2f a = {0.25f, 0.25f};
    v8f c = {};
#pragma unroll
    for (int nt = 0; nt < 4; ++nt) {
        const float bv = b[nt * 16 + nsub];
        v2f bb = {bv, bv};
        v8f d = __builtin_amdgcn_wmma_f32_16x16x4_f32(
            /*neg_a=*/false, a, /*neg_b=*/false, bb,
            /*c_mod=*/(short)0, c, /*reuse_a=*/false, /*reuse_b=*/false);

        float* p = out + (size_t)rbase * OUTC + (nt * 16 + nsub);
#pragma unroll
        for (int v = 0; v < 8; ++v) p[(size_t)v * OUTC] = d[v];
    }
}

// Scalar tail for the last n%16 rows (launched only if needed).
__global__ void __launch_bounds__(256)
link_init_bias_tail(const float* __restrict__ b, float* __restrict__ out,
                    int row0, int nrows) {
    const int nelem = (nrows - row0) * OUTC;
    for (int t = (int)threadIdx.x; t < nelem; t += 256)
        out[(size_t)row0 * OUTC + t] = b[t & (OUTC - 1)];
}

// ---------------------------------------------------------------------------
// K3: one edge per wave32 iteration. 256B coalesced gather of WT[col],
//     64 x atomic f32 add into out[row]. NT index loads; prefetch next column.
// ---------------------------------------------------------------------------
__global__ void __launch_bounds__(256)
link_scatter_wt(const int* __restrict__ rows, const int* __restrict__ cols,
                const float* __restrict__ WT, float* __restrict__ out,
                int nedges) {
    const int lane   = (int)(threadIdx.x & 31);
    const int wave   = (int)((blockIdx.x * blockDim.x + threadIdx.x) >> 5);
    const int nwaves = (int)((gridDim.x * blockDim.x) >> 5);
    const int lo     = lane * 2;

    for (int e = wave; e < nedges; e += nwaves) {
        const int r = __builtin_nontemporal_load(&rows[e]);
        const int c = __builtin_nontemporal_load(&cols[e]);

        const int e2 = e + nwaves;
        if (e2 < nedges) {
            const int c2 = __builtin_nontemporal_load(&cols[e2]);
            __builtin_prefetch(WT + (size_t)c2 * OUTC + lo, 0, 1); // global_prefetch_b8
        }

        const float2 v = *(const float2*)(WT + (size_t)c * OUTC + lo);
        float* dst = out + (size_t)r * OUTC + lo;
        __hip_atomic_fetch_add(dst + 0, v.x, __ATOMIC_RELAXED, __HIP_MEMORY_SCOPE_AGENT);
        __hip_atomic_fetch_add(dst + 1, v.y, __ATOMIC_RELAXED, __HIP_MEMORY_SCOPE_AGENT);
    }
}

// Fallback when d_ws cannot hold WT: strided gather directly from W [OUT, N].
__global__ void __launch_bounds__(256)
link_scatter_direct(const int* __restrict__ rows, const int* __restrict__ cols,
                    const float* __restrict__ W, float* __restrict__ out,
                    int nedges, int n) {
    const int lane   = (int)(threadIdx.x & 31);
    const int wave   = (int)((blockIdx.x * blockDim.x + threadIdx.x) >> 5);
    const int nwaves = (int)((gridDim.x * blockDim.x) >> 5);
    const int o0     = lane * 2;

    for (int e = wave; e < nedges; e += nwaves) {
        const int r = __builtin_nontemporal_load(&rows[e]);
        const int c = __builtin_nontemporal_load(&cols[e]);
        const float v0 = W[(size_t)o0 * (size_t)n + (size_t)c];
        const float v1 = W[(size_t)(o0 + 1) * (size_t)n + (size_t)c];
        float* dst = out + (size_t)r * OUTC + o0;
        __hip_atomic_fetch_add(dst + 0, v0, __ATOMIC_RELAXED, __HIP_MEMORY_SCOPE_AGENT);
        __hip_atomic_fetch_add(dst + 1, v1, __ATOMIC_RELAXED, __HIP_MEMORY_SCOPE_AGENT);
    }
}

extern "C" void kernel_launch(void* const* d_in, const int* in_sizes, int n_in,
                              void* d_out, int out_size, void* d_ws, size_t ws_size,
                              hipStream_t stream) {
    const int*   edge = (const int*)d_in[0];    // [2, E] int32 (harness ABI)
    const float* W    = (const float*)d_in[1];  // [OUT=64, N]
    const float* b    = (const float*)d_in[2];  // [OUT=64]
    float*       out  = (float*)d_out;          // [N, 64]

    const int nE = in_sizes[0] / 2;             // E
    const int n  = in_sizes[1] / OUTC;          // N
    const int* rows = edge;
    const int* cols = edge + nE;

    // K2: out = bias (WMMA broadcast). One wave per 16 rows; 8 waves/block.
    {
        const int ntiles = n / 16;              // full tiles (6250 for N=100000)
        if (ntiles > 0) {
            const int blocks = (ntiles + 7) / 8;
            link_init_bias_wmma<<<blocks, 256, 0, stream>>>(b, out, ntiles);
        }
        const int row0 = ntiles * 16;
        if (row0 < n)                           // ragged tail (none for N=100000)
            link_init_bias_tail<<<1, 256, 0, stream>>>(b, out, row0, n);
    }

    const size_t wt_bytes = (size_t)n * OUTC * sizeof(float);
    if (ws_size >= wt_bytes) {
        float* WT = (float*)d_ws;
        link_transpose_w<<<(n + 63) / 64, 256, 0, stream>>>(W, WT, n);
        // 2048 blocks * 8 waves = 16384 waves; ~195 edges each.
        link_scatter_wt<<<2048, 256, 0, stream>>>(rows, cols, WT, out, nE);
    } else {
        link_scatter_direct<<<2048, 256, 0, stream>>>(rows, cols, W, out, nE, n);
    }
}